// BalanceBCELoss_83923660963951
// MI455X (gfx1250) — compile-verified
//
#include <hip/hip_runtime.h>

// ---------------------------------------------------------------------------
// BalanceBCELoss for MI455X (gfx1250).
// 3 streaming passes (radix-select 12+12+8 bits on loss f32 bit pattern)
// + 3 tiny scan kernels. Streaming uses the CDNA5 Tensor Data Mover to
// double-buffer tiles into LDS (TENSORcnt-synchronized), wave32 WMMA-assisted
// reductions, LDS histograms, hardware float atomics.
// ---------------------------------------------------------------------------

typedef unsigned int u32;
typedef __attribute__((ext_vector_type(4))) u32   v4u;
typedef __attribute__((ext_vector_type(8))) int   v8i;
typedef __attribute__((ext_vector_type(4))) int   v4i;
typedef __attribute__((ext_vector_type(2))) float v2f;
typedef __attribute__((ext_vector_type(8))) float v8f;

#define TILE      1024
#define NTHREADS  256
#define GRID_DIM  2048
#define BINS_HI   4096
#define BINS_LO   256

#ifndef __has_builtin
#define __has_builtin(x) 0
#endif
#if __has_builtin(__builtin_amdgcn_tensor_load_to_lds) && __has_builtin(__builtin_amdgcn_s_wait_tensorcnt)
#define HAVE_TDM 1
#else
#define HAVE_TDM 0
#endif
#if __has_builtin(__builtin_amdgcn_wmma_f32_16x16x4_f32)
#define HAVE_WMMA 1
#else
#define HAVE_WMMA 0
#endif

struct WS {
  u32   cnt0[BINS_HI]; float sum0[BINS_HI];
  u32   cnt1[BINS_HI]; float sum1[BINS_HI];
  u32   cnt2[BINS_LO]; float sum2[BINS_LO];
  float pos_loss, tot_loss, pos_cnt, neg_cnt;
  float sum_above, neg_loss, kf;
  u32   k_rem, prefix, done;
};

__device__ __forceinline__ u32 umin32(u32 a, u32 b) { return a < b ? a : b; }

// Full-wave (32 lane) sum. One V_WMMA_F32_16X16X4_F32 with A=ones folds lane l
// with lane l+16 (column sums of B land in every row of D, lane l of D-VGPR0
// holds s_{l%16}), then 4 xor-shuffles finish the 16-wide reduction.
__device__ __forceinline__ float wave_reduce_sum(float v) {
#if HAVE_WMMA
  v2f a; a[0] = 1.0f; a[1] = 1.0f;   // A = ones(16x4)
  v2f b; b[0] = v;    b[1] = 0.0f;   // B vgpr0 = lane partial, vgpr1 = 0
  v8f c = {};
  v8f d = __builtin_amdgcn_wmma_f32_16x16x4_f32(false, a, false, b, (short)0, c, false, false);
  float s = d[0];                    // s_{lane & 15} in every lane
  s += __shfl_xor(s, 8, 32);
  s += __shfl_xor(s, 4, 32);
  s += __shfl_xor(s, 2, 32);
  s += __shfl_xor(s, 1, 32);
  return s;
#else
  float s = v;
  for (int off = 16; off; off >>= 1) s += __shfl_xor(s, off, 32);
  return s;
#endif
}

#if HAVE_TDM
// Issue one 1-D TDM load: nelem f32/i32 elements from global -> LDS.
// D# group0/group1 built per CDNA5 ISA ch.8 (count=1, type=2, data_size=4B,
// tensor_dim0 = tile_dim0 = nelem, tensor_dim1 = tile_dim1 = 1).
// Toolchain here (clang-23 / therock-10.0) uses the 6-operand builtin:
//   (v4u g0, v8i g1, v4i g2, v4i g3, v8i extra, i32 cpol)
// Groups 2/3 (dims 2..4) and the trailing group are unused for a 2-D tile.
__device__ __forceinline__ void tdm_load_1d(u32 lds_byte_addr, const void* gptr, u32 nelem) {
  unsigned long long ga = (unsigned long long)(uintptr_t)gptr;
  v4u g0;
  g0[0] = 1u;                                        // count=1 (valid), user mode
  g0[1] = lds_byte_addr;                             // lds_addr
  g0[2] = (u32)ga;                                   // global_addr[31:0]
  g0[3] = (u32)((ga >> 32) & 0x01FFFFFFull) | (2u << 30); // global_addr[56:32] | type=2
  v8i g1;
  g1[0] = (int)(2u << 16);                           // data_size=2 (4 bytes)
  g1[1] = (int)((nelem & 0xFFFFu) << 16);            // tensor_dim0[15:0]
  g1[2] = (int)(((nelem >> 16) & 0xFFFFu) | (1u << 16)); // tensor_dim0[31:16] | tensor_dim1=1
  g1[3] = (int)((nelem & 0xFFFFu) << 16);            // tile_dim0 = nelem
  g1[4] = 1;                                         // tile_dim1 = 1
  g1[5] = (int)nelem;                                // tensor_dim0_stride
  g1[6] = 0;
  g1[7] = 0;
  v4i z4 = {0, 0, 0, 0};
  v8i z8 = {0, 0, 0, 0, 0, 0, 0, 0};
  __builtin_amdgcn_tensor_load_to_lds(g0, g1, z4, z4, z8, 0);
}
#endif

// Per-element loss + accumulation. loss = fmax(masked loss, 0) normalizes -0
// so the non-negative float bit pattern is a monotonic sort key.
template <int LEVEL>
__device__ __forceinline__ void elem_update(float p, int tg, float m, u32 prefix,
                                            float& aPosL, float& aTotL,
                                            float& aPosC, float& aNegC,
                                            u32* hcnt, float* hsum) {
  float tb   = (tg == 0) ? 1.0f : 0.0f;
  float t    = tb * m;
  float lp   = fmaxf(__logf(p), -100.0f);
  float l1p  = fmaxf(log1pf(-p), -100.0f);
  float loss = -(t * lp + (1.0f - t) * l1p);
  loss = fmaxf(loss * m, 0.0f);
  if (LEVEL == 0) {
    aTotL += loss;
    if (t == 1.0f) { aPosL += loss; aPosC += 1.0f; }
  }
  if (t == 0.0f) {
    u32 key = __float_as_uint(loss);
    if (LEVEL == 0) {
      aNegC += 1.0f;
      u32 b = key >> 20;
      atomicAdd(&hcnt[b], 1u);
      atomicAdd(&hsum[b], loss);
    } else if (LEVEL == 1) {
      if ((key >> 20) == prefix) {
        u32 b = (key >> 8) & 0xFFFu;
        atomicAdd(&hcnt[b], 1u);
        atomicAdd(&hsum[b], loss);
      }
    } else {
      if ((key >> 8) == prefix) {
        u32 b = key & 0xFFu;
        atomicAdd(&hcnt[b], 1u);
        atomicAdd(&hsum[b], loss);
      }
    }
  }
}

template <int LEVEL>
__global__ __launch_bounds__(NTHREADS)
void stream_pass(const float* __restrict__ pred, const int* __restrict__ targ,
                 const float* __restrict__ mask, WS* __restrict__ ws, u32 n) {
  constexpr int NB = (LEVEL == 2) ? BINS_LO : BINS_HI;
  __shared__ u32   hcnt[NB];
  __shared__ float hsum[NB];
#if HAVE_TDM
  __shared__ float s_pred[2][TILE];
  __shared__ int   s_targ[2][TILE];
  __shared__ float s_mask[2][TILE];
#endif
  const int tid = threadIdx.x;
  for (int i = tid; i < NB; i += NTHREADS) { hcnt[i] = 0u; hsum[i] = 0.0f; }
  u32 prefix = 0;
  if (LEVEL > 0) prefix = ws->prefix;

  float aPosL = 0.f, aTotL = 0.f, aPosC = 0.f, aNegC = 0.f;
  const u32 ntiles = (n + TILE - 1) / TILE;

#if HAVE_TDM
  // Scalar-uniform predicate: TDM ignores EXEC, so waves 1..7 must branch
  // around the issue entirely (s_cbranch on a readfirstlane'd value).
  const bool wave0 = (__builtin_amdgcn_readfirstlane((int)threadIdx.x) < 32);
  u32 t0 = blockIdx.x;
  if (wave0 && t0 < ntiles) {
    u32 base = t0 * TILE, rem = umin32(TILE, n - base);
    tdm_load_1d((u32)(uintptr_t)&s_pred[0][0], pred + base, rem);
    tdm_load_1d((u32)(uintptr_t)&s_targ[0][0], targ + base, rem);
    tdm_load_1d((u32)(uintptr_t)&s_mask[0][0], mask + base, rem);
  }
  int buf = 0;
  for (u32 t = t0; t < ntiles; t += gridDim.x) {
    u32 nxt = t + gridDim.x;
    if (wave0) {
      if (nxt < ntiles) {
        u32 nb = nxt * TILE, nrem = umin32(TILE, n - nb);
        int nbuf = buf ^ 1;
        tdm_load_1d((u32)(uintptr_t)&s_pred[nbuf][0], pred + nb, nrem);
        tdm_load_1d((u32)(uintptr_t)&s_targ[nbuf][0], targ + nb, nrem);
        tdm_load_1d((u32)(uintptr_t)&s_mask[nbuf][0], mask + nb, nrem);
        __builtin_amdgcn_s_wait_tensorcnt(3);   // current tile's 3 loads done
      } else {
        __builtin_amdgcn_s_wait_tensorcnt(0);
      }
    }
    __syncthreads();
    u32 base = t * TILE, rem = umin32(TILE, n - base);
#pragma unroll
    for (int j = 0; j < TILE / NTHREADS; ++j) {
      u32 i = (u32)tid + (u32)j * NTHREADS;
      if (i < rem)
        elem_update<LEVEL>(s_pred[buf][i], s_targ[buf][i], s_mask[buf][i], prefix,
                           aPosL, aTotL, aPosC, aNegC, hcnt, hsum);
    }
    __syncthreads();
    buf ^= 1;
  }
#else
  __syncthreads();
  for (u32 t = blockIdx.x; t < ntiles; t += gridDim.x) {
    u32 base = t * TILE, rem = umin32(TILE, n - base);
#pragma unroll
    for (int j = 0; j < TILE / NTHREADS; ++j) {
      u32 i = (u32)tid + (u32)j * NTHREADS;
      if (i < rem) {
        size_t g = (size_t)base + i;
        elem_update<LEVEL>(pred[g], targ[g], mask[g], prefix,
                           aPosL, aTotL, aPosC, aNegC, hcnt, hsum);
      }
    }
  }
#endif

  if (LEVEL == 0) {   // uniform control flow here -> EXEC all ones for WMMA
    float rPosL = wave_reduce_sum(aPosL);
    float rTotL = wave_reduce_sum(aTotL);
    float rPosC = wave_reduce_sum(aPosC);
    float rNegC = wave_reduce_sum(aNegC);
    if ((tid & 31) == 0) {
      unsafeAtomicAdd(&ws->pos_loss, rPosL);
      unsafeAtomicAdd(&ws->tot_loss, rTotL);
      unsafeAtomicAdd(&ws->pos_cnt,  rPosC);
      unsafeAtomicAdd(&ws->neg_cnt,  rNegC);
    }
  }
  __syncthreads();
  u32*   gcnt = (LEVEL == 0) ? ws->cnt0 : (LEVEL == 1 ? ws->cnt1 : ws->cnt2);
  float* gsum = (LEVEL == 0) ? ws->sum0 : (LEVEL == 1 ? ws->sum1 : ws->sum2);
  for (int i = tid; i < NB; i += NTHREADS) {
    u32 c = hcnt[i];
    if (c) {
      atomicAdd(&gcnt[i], c);
      unsafeAtomicAdd(&gsum[i], hsum[i]);
    }
  }
}

__global__ void scan0(WS* ws) {
  float posc = ws->pos_cnt, negc = ws->neg_cnt;
  float kf = fminf(negc, posc * 5.0f);   // f32, same rounding as reference
  ws->kf = kf;
  u32 k = (u32)ceilf(kf);                // #ranks with rank < kf
  if (k == 0u) {
    ws->done = 1u; ws->neg_loss = 0.0f; ws->prefix = 0xFFFFFFFFu;
    ws->k_rem = 0u; ws->sum_above = 0.0f;
    return;
  }
  u32 cum = 0; float sum = 0.0f; int bsel = 0;
  for (int b = BINS_HI - 1; b >= 0; --b) {
    u32 c = ws->cnt0[b];
    if (cum + c >= k) { bsel = b; break; }
    cum += c; sum += ws->sum0[b];
  }
  ws->prefix = (u32)bsel; ws->k_rem = k - cum; ws->sum_above = sum; ws->done = 0u;
}

__global__ void scan1(WS* ws) {
  if (ws->done) { ws->prefix = 0xFFFFFFFFu; return; }
  u32 k = ws->k_rem, cum = 0; float sum = ws->sum_above; int bsel = 0;
  for (int b = BINS_HI - 1; b >= 0; --b) {
    u32 c = ws->cnt1[b];
    if (cum + c >= k) { bsel = b; break; }
    cum += c; sum += ws->sum1[b];
  }
  ws->prefix = (ws->prefix << 12) | (u32)bsel;
  ws->k_rem = k - cum; ws->sum_above = sum;
}

__global__ void scan2(WS* ws, float* out, u32 n) {
  float neg_loss;
  float kf = ws->kf;
  if (ws->done) {
    neg_loss = ws->neg_loss;
  } else {
    u32 k = ws->k_rem, cum = 0; float sum = ws->sum_above; int bsel = 0;
    for (int b = BINS_LO - 1; b >= 0; --b) {
      u32 c = ws->cnt2[b];
      if (cum + c >= k) { bsel = b; break; }
      cum += c; sum += ws->sum2[b];
    }
    u32 tau_bits = (ws->prefix << 8) | (u32)bsel;   // exact 32-bit threshold
    float tau = __uint_as_float(tau_bits);
    neg_loss = sum + (float)(k - cum) * tau;        // ties at tau, like rank mask
  }
  float posc = ws->pos_cnt;
  float res = (posc == 0.0f)
                  ? (ws->tot_loss / (float)n)       // fallback: mean(loss)
                  : (ws->pos_loss + neg_loss) / (posc + kf + 1e-8f);
  out[0] = res;
}

extern "C" void kernel_launch(void* const* d_in, const int* in_sizes, int n_in,
                              void* d_out, int out_size, void* d_ws, size_t ws_size,
                              hipStream_t stream) {
  const float* pred = (const float*)d_in[0];
  const int*   targ = (const int*)d_in[1];
  const float* mask = (const float*)d_in[2];
  float* out = (float*)d_out;
  WS* ws = (WS*)d_ws;
  u32 n = (u32)in_sizes[0];

  (void)hipMemsetAsync(d_ws, 0, sizeof(WS), stream);
  stream_pass<0><<<GRID_DIM, NTHREADS, 0, stream>>>(pred, targ, mask, ws, n);
  scan0<<<1, 1, 0, stream>>>(ws);
  stream_pass<1><<<GRID_DIM, NTHREADS, 0, stream>>>(pred, targ, mask, ws, n);
  scan1<<<1, 1, 0, stream>>>(ws);
  stream_pass<2><<<GRID_DIM, NTHREADS, 0, stream>>>(pred, targ, mask, ws, n);
  scan2<<<1, 1, 0, stream>>>(ws, out, n);
}